// RoIPooling2D_13709535608927
// MI455X (gfx1250) — compile-verified
//
#include <hip/hip_runtime.h>
#include <stdint.h>

typedef unsigned int v4u __attribute__((ext_vector_type(4)));
typedef int          v8i __attribute__((ext_vector_type(8)));
typedef int          v4i __attribute__((ext_vector_type(4)));

#define CB     4          // channels per block (tile z-dim)
#define C_TOT  256
#define HH     50
#define WW     50
#define PHW    49         // 7x7 bins
#define LDS_BYTES (HH * WW * CB * 4)   // 40000 B worst-case tile, < 64KB

#if __has_builtin(__builtin_amdgcn_readfirstlane)
#define RFL(v) ((unsigned int)__builtin_amdgcn_readfirstlane((int)(v)))
#else
#define RFL(v) ((unsigned int)(v))
#endif

__global__ __launch_bounds__(256) void roipool_tdm_kernel(
    const float* __restrict__ x, const float* __restrict__ rois,
    float* __restrict__ out)
{
  extern __shared__ float tile[];   // packed [CB][th][tw]

  const int blk = blockIdx.x;
  const int r   = blk >> 6;            // 64 channel-blocks per roi
  const int c0  = (blk & 63) * CB;

  // ---- block-uniform ROI parameters (matches reference exactly) ----
  const float* rp = rois + r * 5;
  const int b  = (int)rp[0];
  const int xs = (int)rintf(rp[1] * 0.0625f);  // v_rndne == jnp.round (half-to-even)
  const int ys = (int)rintf(rp[2] * 0.0625f);
  const int xe = (int)rintf(rp[3] * 0.0625f);
  const int ye = (int)rintf(rp[4] * 0.0625f);
  const float rw = (float)max(xe - xs + 1, 1);
  const float rh = (float)max(ye - ys + 1, 1);
  const float bh = rh / 7.0f;
  const float bw = rw / 7.0f;

  // union of all bin ranges for this roi (bounds of ph=0 start / ph=6 end)
  const int row0 = min(max(ys, 0), HH);
  const int row1 = min(max((int)ceilf(7.0f * bh) + ys, 0), HH);
  const int col0 = min(max(xs, 0), WW);
  const int col1 = min(max((int)ceilf(7.0f * bw) + xs, 0), WW);
  const int th = row1 - row0;
  const int tw = col1 - col0;

  // ---- stage tile [CB][th][tw] into LDS via the Tensor Data Mover ----
#if __has_builtin(__builtin_amdgcn_tensor_load_to_lds)
  if (th > 0 && tw > 0 && threadIdx.x < 32) {   // wave 0 only issues the TDM op
    const uint64_t gaddr = (uint64_t)(uintptr_t)x +
        4ull * (uint64_t)(((b * C_TOT + c0) * (HH * WW)) + row0 * WW + col0);
    const uint32_t lds_base = (uint32_t)(uintptr_t)&tile[0]; // addr[31:0] = LDS offset

    v4u g0;
    g0[0] = RFL(1u);                                    // count=1 (valid user D#)
    g0[1] = RFL(lds_base);                              // lds_addr
    g0[2] = RFL((uint32_t)gaddr);                       // global_addr[31:0]
    g0[3] = RFL(((uint32_t)(gaddr >> 32) & 0x1FFFFFFu)  // global_addr[56:32]
                | (2u << 30));                          // type=2 ("image")

    v8i g1;
    g1[0] = (int)RFL(2u << 16);                         // data_size=4B, no multicast
    g1[1] = (int)RFL((uint32_t)tw << 16);               // tensor_dim0 = tw
    g1[2] = (int)RFL((uint32_t)th << 16);               // tensor_dim1 = th
    g1[3] = (int)RFL((uint32_t)tw << 16);               // tile_dim0   = tw
    g1[4] = (int)RFL(((uint32_t)CB << 16) | (uint32_t)th); // tile_dim1=th, tile_dim2=CB
    g1[5] = (int)RFL((uint32_t)WW);                     // tensor_dim0_stride = 50
    g1[6] = (int)RFL((uint32_t)(HH * WW) << 16);        // tensor_dim1_stride lo16 = 2500
    g1[7] = (int)RFL((uint32_t)(HH * WW) >> 16);        // tensor_dim1_stride hi = 0

    v4i g2;
    g2[0] = (int)RFL((uint32_t)CB);                     // tensor_dim2 = CB
    g2[1] = 0; g2[2] = 0; g2[3] = 0;
    v4i g3 = {0, 0, 0, 0};
    v8i g4 = {0, 0, 0, 0, 0, 0, 0, 0};                  // extra operand (clang-23 6-arg form)

    __builtin_amdgcn_tensor_load_to_lds(g0, g1, g2, g3, g4, /*cpol=*/0);
#if __has_builtin(__builtin_amdgcn_s_wait_tensorcnt)
    __builtin_amdgcn_s_wait_tensorcnt(0);
#else
    asm volatile("s_wait_tensorcnt 0x0" ::: "memory");
#endif
  }
#else
  // Fallback: cooperative global->LDS copy (no TDM builtin on this toolchain)
  if (th > 0 && tw > 0) {
    const int total = th * tw * CB;
    for (int i = threadIdx.x; i < total; i += blockDim.x) {
      const int cl  = i / (th * tw);
      const int rem = i - cl * th * tw;
      const int hh  = rem / tw;
      const int wcol = rem - hh * tw;
      tile[i] = x[((b * C_TOT + c0 + cl) * HH + row0 + hh) * WW + col0 + wcol];
    }
  }
#endif
  __syncthreads();

  // ---- each thread: one (channel, bin) max-reduction from LDS ----
  const int work = threadIdx.x;
  if (work < CB * PHW) {
    const int cl  = work / PHW;
    const int bin = work - cl * PHW;
    const int ph  = bin / 7;
    const int pw  = bin - ph * 7;

    const int hs = min(max((int)floorf((float)ph * bh) + ys, 0), HH);
    const int he = min(max((int)ceilf((float)(ph + 1) * bh) + ys, 0), HH);
    const int ws = min(max((int)floorf((float)pw * bw) + xs, 0), WW);
    const int we = min(max((int)ceilf((float)(pw + 1) * bw) + xs, 0), WW);

    float m = -INFINITY;
    const float* base = tile + (size_t)cl * th * tw;
    for (int h = hs; h < he; ++h) {
      const float* rowp = base + (size_t)(h - row0) * tw - col0;
      for (int w = ws; w < we; ++w) m = fmaxf(m, rowp[w]);
    }
    const bool valid = (hs < he) && (ws < we);
    out[(size_t)(r * C_TOT + c0 + cl) * PHW + bin] = valid ? m : 0.0f;
  }
}

extern "C" void kernel_launch(void* const* d_in, const int* in_sizes, int n_in,
                              void* d_out, int out_size, void* d_ws, size_t ws_size,
                              hipStream_t stream) {
  const float* x    = (const float*)d_in[0];
  const float* rois = (const float*)d_in[1];
  float* out        = (float*)d_out;

  const int nroi = in_sizes[1] / 5;                 // 1024
  const int nblk = nroi * (C_TOT / CB);             // 65536 blocks
  roipool_tdm_kernel<<<nblk, 256, LDS_BYTES, stream>>>(x, rois, out);
}